// CompetitiveLayer_1_6279242186886
// MI455X (gfx1250) — compile-verified
//
#include <hip/hip_runtime.h>

typedef __attribute__((ext_vector_type(2))) float v2f;
typedef __attribute__((ext_vector_type(4))) float v4f;
typedef __attribute__((ext_vector_type(8))) float v8f;

#define N 4096

// ---------------------------------------------------------------------------
// AF[i] = AT[i] / (1 + sum_j K[i][j] * BF[j]),  K = sqrt_K^2 (squared on load)
// Row reduction via V_WMMA_F32_16X16X4_F32: A = 16x4 K-tile, B = BF broadcast
// across all 16 N columns, so every column of D holds the 16 row partial sums.
// Block = 8 waves; block b owns rows [16b,16b+16); wave w reduces j in
// [512w, 512w+512); LDS combines the 8 wave partials.
// ---------------------------------------------------------------------------
__global__ void __launch_bounds__(256) af_kernel(const float* __restrict__ sq,
                                                 const float* __restrict__ AT,
                                                 const float* __restrict__ BF,
                                                 float* __restrict__ AF) {
  __shared__ float red[8][16];
  const int wave = threadIdx.x >> 5;
  const int lane = threadIdx.x & 31;
  const int i0   = blockIdx.x << 4;          // 16 rows per block
  const int row  = i0 + (lane & 15);
  const int koff = (lane >> 4) << 1;         // lanes 0-15 -> K=0,1 ; 16-31 -> K=2,3
  const float* arow = sq + (size_t)row * N;

  v8f c = {};
  const int jbeg = wave << 9;                // 512 columns per wave
#pragma unroll 4
  for (int j = jbeg; j < jbeg + 512; j += 4) {
    v2f s = *(const v2f*)(arow + j + koff);  // sqrt_K pair for this lane's K slots
    v2f a; a.x = s.x * s.x; a.y = s.y * s.y; // K = sqrt_K^2, squared in-register
    v2f b = *(const v2f*)(BF + j + koff);    // BF broadcast across all N columns
    c = __builtin_amdgcn_wmma_f32_16x16x4_f32(false, a, false, b,
                                              (short)0, c, false, false);
  }
  // D layout: lane n(0-15), VGPR m -> [M=m][N=n]; lanes 16-31 -> M=8+m.
  // All N columns identical -> lane 0 / lane 16 carry the 16 row sums.
  if (lane == 0) {
#pragma unroll
    for (int m = 0; m < 8; ++m) red[wave][m] = c[m];
  } else if (lane == 16) {
#pragma unroll
    for (int m = 0; m < 8; ++m) red[wave][8 + m] = c[m];
  }
  __syncthreads();
  if (threadIdx.x < 16) {
    float s = 0.f;
#pragma unroll
    for (int w = 0; w < 8; ++w) s += red[w][threadIdx.x];
    AF[i0 + threadIdx.x] = AT[i0 + threadIdx.x] / (1.0f + s);
  }
}

// ---------------------------------------------------------------------------
// BF[j] = BT[j] / (1 + sum_i AF[i] * K[i][j])
// Column reduction: B = 4x16 K-tile (coalesced 64B row segments across lanes
// 0-15), A = AF[i..i+3] broadcast across all 16 M rows, so D row 0 holds the
// 16 column partial sums.
// ---------------------------------------------------------------------------
__global__ void __launch_bounds__(256) bf_kernel(const float* __restrict__ sq,
                                                 const float* __restrict__ BT,
                                                 const float* __restrict__ AF,
                                                 float* __restrict__ BF) {
  __shared__ float red[8][16];
  const int wave = threadIdx.x >> 5;
  const int lane = threadIdx.x & 31;
  const int j0   = blockIdx.x << 4;          // 16 columns per block
  const int col  = j0 + (lane & 15);
  const int koff = (lane >> 4) << 1;         // row offset within 4-row tile

  v8f c = {};
  const int ibeg = wave << 9;                // 512 rows per wave
#pragma unroll 4
  for (int i = ibeg; i < ibeg + 512; i += 4) {
    const float* p = sq + (size_t)(i + koff) * N + col;
    float s0 = p[0];                         // K[i+koff  ][col]
    float s1 = p[N];                         // K[i+koff+1][col]
    v2f b; b.x = s0 * s0; b.y = s1 * s1;
    v2f a = *(const v2f*)(AF + i + koff);    // AF broadcast across all M rows
    c = __builtin_amdgcn_wmma_f32_16x16x4_f32(false, a, false, b,
                                              (short)0, c, false, false);
  }
  // All M rows identical -> lane n (0-15), VGPR 0 = column sum for j0+n.
  if (lane < 16) red[wave][lane] = c[0];
  __syncthreads();
  if (threadIdx.x < 16) {
    float s = 0.f;
#pragma unroll
    for (int w = 0; w < 8; ++w) s += red[w][threadIdx.x];
    BF[j0 + threadIdx.x] = BT[j0 + threadIdx.x] / (1.0f + s);
  }
}

// ---------------------------------------------------------------------------
// C[i][j] = sqrt_K[i][j]^2 * AF[i] * BF[j]   (elementwise, float4 streaming;
// non-temporal store so the 64 MB output does not evict K from L2)
// ---------------------------------------------------------------------------
__global__ void __launch_bounds__(256) c_kernel(const float* __restrict__ sq,
                                                const float* __restrict__ AF,
                                                const float* __restrict__ BF,
                                                float* __restrict__ C) {
  const size_t t    = (size_t)blockIdx.x * 256 + threadIdx.x;
  const size_t base = t * 4;                 // 4 | N -> row uniform per float4
  const int row = (int)(base >> 12);
  const int col = (int)(base & (N - 1));
  v4f s = *(const v4f*)(sq + base);
  v4f b = *(const v4f*)(BF + col);
  const float af = AF[row];
  v4f o;
  o.x = s.x * s.x * af * b.x;
  o.y = s.y * s.y * af * b.y;
  o.z = s.z * s.z * af * b.z;
  o.w = s.w * s.w * af * b.w;
  __builtin_nontemporal_store(o, (v4f*)(C + base));
}

extern "C" void kernel_launch(void* const* d_in, const int* in_sizes, int n_in,
                              void* d_out, int out_size, void* d_ws, size_t ws_size,
                              hipStream_t stream) {
  (void)in_sizes; (void)n_in; (void)out_size; (void)ws_size;
  const float* AT = (const float*)d_in[0];
  const float* BT = (const float*)d_in[1];
  const float* SQ = (const float*)d_in[2];   // sqrt_K, 4096x4096 row-major
  float* AF = (float*)d_ws;                  // 4096 floats
  float* BF = AF + N;                        // 4096 floats
  float* C  = (float*)d_out;

  const dim3 blk(256);
  // Iteration 0 seeds the B-side with BT (reference carry starts at (AT, BT)).
  af_kernel<<<N / 16, blk, 0, stream>>>(SQ, AT, BT, AF);
  bf_kernel<<<N / 16, blk, 0, stream>>>(SQ, BT, AF, BF);
  for (int it = 1; it < 64; ++it) {
    af_kernel<<<N / 16, blk, 0, stream>>>(SQ, AT, BF, AF);
    bf_kernel<<<N / 16, blk, 0, stream>>>(SQ, BT, AF, BF);
  }
  c_kernel<<<(N * (size_t)N / 4) / 256, blk, 0, stream>>>(SQ, AF, BF, C);
}